// HybridQuantumClassicalModel_71459665871630
// MI455X (gfx1250) — compile-verified
//
#include <hip/hip_runtime.h>
#include <math.h>

typedef __attribute__((ext_vector_type(2))) float v2f;
typedef __attribute__((ext_vector_type(8))) float v8f;

#define WAVES   8
#define ITERS   8
#define HSTRIDE 34   // 16x32 tile with stride 34: conflict-free stores, 8B-aligned b64 reloads

__device__ __forceinline__ v8f wmma4(v2f a, v2f b, v8f c) {
    // V_WMMA_F32_16X16X4_F32: D(16x16,f32) = A(16x4,f32) * B(4x16,f32) + C
    return __builtin_amdgcn_wmma_f32_16x16x4_f32(false, a, false, b, (short)0, c, false, false);
}

// ---------------------------------------------------------------------------
// Quantum head: 8-qubit statevector sim (256 complex amps), both circuits.
// One block of 256 threads; thread t owns amplitude index t. Writes
// ws[0] = coeffs_amplitude @ za, ws[1] = coeffs_phase @ zp.
// ---------------------------------------------------------------------------
__global__ __launch_bounds__(256) void quantum_head_kernel(
    const float* __restrict__ pa, const float* __restrict__ pp,
    const float* __restrict__ ca, const float* __restrict__ cp,
    float* __restrict__ res)
{
    __shared__ float sre[256], sim_[256];
    __shared__ float red[8];
    const int t = threadIdx.x;

    for (int circ = 0; circ < 2; ++circ) {
        const float* p  = circ ? pp : pa;
        const float* cf = circ ? cp : ca;
        sre[t] = 0.0625f;   // 2^(-8/2): uniform state after Hadamards
        sim_[t] = 0.0f;
        if (t < 8) red[t] = 0.0f;
        __syncthreads();

        for (int layer = 0; layer < 4; ++layer) {
            // fused single-qubit rotations: U = Ry * Rx * Rz
            for (int q = 0; q < 8; ++q) {
                const float thz = p[(layer * 8 + q) * 3 + 0];
                const float thx = p[(layer * 8 + q) * 3 + 1];
                const float thy = p[(layer * 8 + q) * 3 + 2];
                const float cz = cosf(0.5f * thz), sz = sinf(0.5f * thz);
                const float cx = cosf(0.5f * thx), sx = sinf(0.5f * thx);
                const float cy = cosf(0.5f * thy), sy = sinf(0.5f * thy);
                // M = Rx * Rz
                const float M00r =  cx * cz, M00i = -cx * sz;
                const float M01r =  sx * sz, M01i = -sx * cz;
                const float M10r = -sx * sz, M10i = -sx * cz;
                const float M11r =  cx * cz, M11i =  cx * sz;
                // U = Ry * M
                const float U00r = cy * M00r - sy * M10r, U00i = cy * M00i - sy * M10i;
                const float U01r = cy * M01r - sy * M11r, U01i = cy * M01i - sy * M11i;
                const float U10r = sy * M00r + cy * M10r, U10i = sy * M00i + cy * M10i;
                const float U11r = sy * M01r + cy * M11r, U11i = sy * M01i + cy * M11i;
                const int bit = 7 - q;            // axis q -> bit (7-q) of flat index
                if (t < 128) {
                    const int mask = (1 << bit) - 1;
                    const int i0 = ((t & ~mask) << 1) | (t & mask);
                    const int i1 = i0 | (1 << bit);
                    const float a0r = sre[i0], a0i = sim_[i0];
                    const float a1r = sre[i1], a1i = sim_[i1];
                    sre[i0]  = U00r * a0r - U00i * a0i + U01r * a1r - U01i * a1i;
                    sim_[i0] = U00r * a0i + U00i * a0r + U01r * a1i + U01i * a1r;
                    sre[i1]  = U10r * a0r - U10i * a0i + U11r * a1r - U11i * a1i;
                    sim_[i1] = U10r * a0i + U10i * a0r + U11r * a1i + U11i * a1r;
                }
                __syncthreads();
            }
            // entangling layer: CNOT if (i+j) even else CZ, wires (i=control, j=target)
            for (int i = 0; i < 8; ++i) {
                for (int j = i + 1; j < 8; ++j) {
                    const int bc = 7 - i, bt = 7 - j;
                    const int ctrl = (t >> bc) & 1;
                    const int tgt  = (t >> bt) & 1;
                    if (((i + j) & 1) == 0) {
                        // CNOT: new[t] = ctrl ? old[t ^ tbit] : old[t]
                        const int src = ctrl ? (t ^ (1 << bt)) : t;
                        const float r = sre[src], im = sim_[src];
                        __syncthreads();
                        sre[t] = r; sim_[t] = im;
                        __syncthreads();
                    } else {
                        // CZ: sign flip where both bits set (own element only)
                        if (ctrl & tgt) { sre[t] = -sre[t]; sim_[t] = -sim_[t]; }
                        __syncthreads();
                    }
                }
            }
        }
        // PauliZ expvals and coefficient dot
        const float prob = sre[t] * sre[t] + sim_[t] * sim_[t];
        for (int q = 0; q < 8; ++q) {
            const float sgn = ((t >> (7 - q)) & 1) ? -1.0f : 1.0f;
            atomicAdd(&red[q], prob * sgn);
        }
        __syncthreads();
        if (t == 0) {
            float acc = 0.0f;
            for (int q = 0; q < 8; ++q) acc += cf[q] * red[q];
            res[circ] = acc;
        }
        __syncthreads();
    }
}

// ---------------------------------------------------------------------------
// Batched MLP + complex log, WMMA f32 16x16x4. 8 waves/block; each wave owns
// a 16-row tile per iteration.
// ---------------------------------------------------------------------------
__global__ __launch_bounds__(256) void mlp_logpsi_kernel(
    const float* __restrict__ x,
    const float* __restrict__ W1, const float* __restrict__ b1,
    const float* __restrict__ W2, const float* __restrict__ b2,
    const float* __restrict__ W3, const float* __restrict__ b3,
    const float* __restrict__ qres,
    float* __restrict__ out, int B)
{
    __shared__ float lds[WAVES][16 * HSTRIDE];
    const int lane  = threadIdx.x & 31;
    const int wave  = threadIdx.x >> 5;
    const int lrow  = lane & 15;          // M (A-frag) or N (B/D-frag) position
    const int khalf = (lane >> 4) << 1;   // 0 or 2: K-pair selector
    const int mbase = (lane >> 4) << 3;   // 0 or 8: D-layout M offset

    // ---- preload B-matrix fragments (4x16 f32 layout) for W1, W2 ----
    v2f w1f[2][2];                        // [k-chunk][n-tile]
    v2f w2f[8][2];
    for (int kc = 0; kc < 2; ++kc)
        for (int nt = 0; nt < 2; ++nt) {
            const int n  = nt * 16 + lrow;
            const int kk = 4 * kc + khalf;
            v2f f; f[0] = W1[kk * 32 + n]; f[1] = W1[(kk + 1) * 32 + n];
            w1f[kc][nt] = f;
        }
    for (int kc = 0; kc < 8; ++kc)
        for (int nt = 0; nt < 2; ++nt) {
            const int n  = nt * 16 + lrow;
            const int kk = 4 * kc + khalf;
            v2f f; f[0] = W2[kk * 32 + n]; f[1] = W2[(kk + 1) * 32 + n];
            w2f[kc][nt] = f;
        }
    float bias1[2], bias2[2];
    bias1[0] = b1[lrow]; bias1[1] = b1[16 + lrow];
    bias2[0] = b2[lrow]; bias2[1] = b2[16 + lrow];
    const float amp = qres[0];
    const float phs = qres[1];
    const float bb3 = b3[0];

    float* h = lds[wave];
    const int tile0 = (blockIdx.x * WAVES + wave) * ITERS;

    for (int it = 0; it < ITERS; ++it) {
        const int row0 = (tile0 + it) * 16;
        if (row0 >= B) break;             // wave-uniform: EXEC stays all-ones

        // ---- layer 1: (16x8) @ (8x32), K split into 2 chunks of 4 ----
        const float* xr = x + (size_t)(row0 + lrow) * 8 + khalf;
        v2f a0; a0[0] = xr[0]; a0[1] = xr[1];
        v2f a1; a1[0] = xr[4]; a1[1] = xr[5];
        v8f acc1[2];
        for (int nt = 0; nt < 2; ++nt) {
            v8f cc;
            for (int i = 0; i < 8; ++i) cc[i] = bias1[nt];   // bias folded into C
            cc = wmma4(a0, w1f[0][nt], cc);
            cc = wmma4(a1, w1f[1][nt], cc);
            acc1[nt] = cc;
        }
        // relu -> LDS (D-layout scatter, conflict-free with stride 34)
        for (int i = 0; i < 8; ++i) {
            const int m = mbase + i;
            h[m * HSTRIDE + lrow]      = fmaxf(acc1[0][i], 0.0f);
            h[m * HSTRIDE + 16 + lrow] = fmaxf(acc1[1][i], 0.0f);
        }
        asm volatile("s_wait_dscnt 0" ::: "memory");

        // ---- layer 2: (16x32) @ (32x32), 8 K-chunks ----
        v2f af[8];
        for (int kc = 0; kc < 8; ++kc)
            af[kc] = *(const v2f*)&h[lrow * HSTRIDE + 4 * kc + khalf];
        v8f acc2[2];
        for (int nt = 0; nt < 2; ++nt) {
            v8f cc;
            for (int i = 0; i < 8; ++i) cc[i] = bias2[nt];
            for (int kc = 0; kc < 8; ++kc)
                cc = wmma4(af[kc], w2f[kc][nt], cc);
            acc2[nt] = cc;
        }
        // relu -> LDS again for the 32-wide output dot
        for (int i = 0; i < 8; ++i) {
            const int m = mbase + i;
            h[m * HSTRIDE + lrow]      = fmaxf(acc2[0][i], 0.0f);
            h[m * HSTRIDE + 16 + lrow] = fmaxf(acc2[1][i], 0.0f);
        }
        asm volatile("s_wait_dscnt 0" ::: "memory");

        // ---- layer 3 + complex log: lanes 0..15 each finish one row ----
        if (lane < 16) {
            float cls = bb3;
            for (int k = 0; k < 32; ++k)
                cls += h[lane * HSTRIDE + k] * W3[k];
            const float re = amp + logf(fabsf(cls));
            float im = phs;
            if (cls < 0.0f) im += 3.14159265358979323846f;
            if (im > 3.14159265358979323846f)       im -= 6.283185307179586f;
            else if (im <= -3.14159265358979323846f) im += 6.283185307179586f;
            const size_t r = (size_t)row0 + lane;
            out[2 * r]     = re;   // complex64 flattened (re, im)
            out[2 * r + 1] = im;
        }
    }
}

extern "C" void kernel_launch(void* const* d_in, const int* in_sizes, int n_in,
                              void* d_out, int out_size, void* d_ws, size_t ws_size,
                              hipStream_t stream) {
    const float* x   = (const float*)d_in[0];
    const float* qpa = (const float*)d_in[1];
    const float* qpp = (const float*)d_in[2];
    const float* ca  = (const float*)d_in[3];
    const float* cp  = (const float*)d_in[4];
    const float* W1  = (const float*)d_in[5];
    const float* b1  = (const float*)d_in[6];
    const float* W2  = (const float*)d_in[7];
    const float* b2  = (const float*)d_in[8];
    const float* W3  = (const float*)d_in[9];
    const float* b3  = (const float*)d_in[10];
    float* qres = (float*)d_ws;          // [0]=amplitude, [1]=phase
    float* out  = (float*)d_out;
    const int B = in_sizes[0] / 8;

    quantum_head_kernel<<<1, 256, 0, stream>>>(qpa, qpp, ca, cp, qres);

    const int rows_per_block = 16 * WAVES * ITERS;   // 1024
    const int blocks = (B + rows_per_block - 1) / rows_per_block;
    mlp_logpsi_kernel<<<blocks, 256, 0, stream>>>(x, W1, b1, W2, b2, W3, b3,
                                                  qres, out, B);
}